// GCNN_13786845020966
// MI455X (gfx1250) — compile-verified
//
#include <hip/hip_runtime.h>

// ---- problem constants (match reference) ----
#define BATCH 8
#define NNODE 10000
#define NEDGE 320000
#define CIN   128
#define COUT  128

typedef __attribute__((ext_vector_type(2))) float v2f;
typedef __attribute__((ext_vector_type(8))) float v8f;

// ------------------------------------------------------------------
// Kernel 0: zero the agg accumulator in workspace (harness poisons ws).
// ------------------------------------------------------------------
__global__ void gcn_zero_ws(float4* __restrict__ ws, int n4) {
    int i = blockIdx.x * blockDim.x + threadIdx.x;
    if (i < n4) ws[i] = make_float4(0.f, 0.f, 0.f, 0.f);
}

// ------------------------------------------------------------------
// Kernel 1: batched COO SpMM via L2-resident fp32 atomics.
// One wave32 per edge: lane l moves x[b, col, 4l..4l+3] (float4,
// 512B coalesced per wave) scaled by val into agg[b, row, :].
// x + agg working set (~82MB) fits in the 192MB L2, so the
// non-returning global_atomic_add_f32 scatter resolves in L2.
// ------------------------------------------------------------------
__global__ void gcn_spmm_atomic(const float* __restrict__ x,
                                const int*   __restrict__ rows,
                                const int*   __restrict__ cols,
                                const float* __restrict__ vals,
                                float*       __restrict__ agg) {
    long gid  = (long)blockIdx.x * blockDim.x + threadIdx.x;
    long edge = gid >> 5;                 // one wave per edge
    int  lane = threadIdx.x & 31;
    if (edge >= (long)BATCH * NEDGE) return;

    int b = (int)(edge / NEDGE);
    int r = rows[edge];
    int c = cols[edge];
    float v = vals[edge];

    const float4 xv =
        ((const float4*)(x + ((size_t)b * NNODE + c) * CIN))[lane];
    float* dst = agg + ((size_t)b * NNODE + r) * CIN + lane * 4;

    atomicAdd(dst + 0, v * xv.x);         // global_atomic_add_f32 (no return)
    atomicAdd(dst + 1, v * xv.y);
    atomicAdd(dst + 2, v * xv.z);
    atomicAdd(dst + 3, v * xv.w);
}

// ------------------------------------------------------------------
// Kernel 2: dense GEMM [80000,128]x[128,128] + bias + ReLU.
//   - W staged transposed in LDS once per block (64KB, XOR-swizzled
//     so B-fragment ds loads are bank-conflict-free b64s).
//   - One wave owns a 16-row slab and computes ALL 8 column tiles:
//     each A fragment feeds 8 v_wmma_f32_16x16x4_f32 ops.
//   - Per K-step: LOAD phase (all 4 ds_load_2addr issue first, forced
//     by a sched_barrier -> 4 live dest quads, staggered dscnt waits)
//     then COMPUTE phase (8 WMMAs with LDS latency hidden behind them).
//
// Fragment layouts (ISA 7.12.2, f32 16x16x4):
//   A 16x4: lanes 0-15 -> M=lane,   K={k,k+1}; lanes 16-31 -> M=lane-16, K={k+2,k+3}
//   B 4x16: lane's N = lane%16, same K-pair selection by lane half
//   C/D   : VGPR m -> M = m + 8*(lane>=16), N = lane%16
// ------------------------------------------------------------------
__global__ void gcn_gemm_wmma(const float* __restrict__ agg,
                              const float* __restrict__ W,
                              const float* __restrict__ bias,
                              float*       __restrict__ out) {
    __shared__ float ldsWT[CIN * COUT];   // 64KB: W transposed + swizzled

    int tid  = threadIdx.x;
    int lane = tid & 31;
    int wv   = tid >> 5;                  // wave id in block (0..7)
    int half = lane >> 4;                 // K-pair selector
    int l16  = lane & 15;

    // ---- stage W^T into LDS, XOR-swizzled: ldsWT[c*128 + (k ^ 4*(c&15))] ----
    // Swizzle keeps even K-pairs contiguous (b64) and spreads the 16
    // lanes of each half-wave across disjoint bank quads.
    for (int idx = tid; idx < CIN * COUT; idx += 256) {
        int k = idx >> 7;                 // W row   (coalesced global read)
        int c = idx & 127;                // W col
        ldsWT[(c << 7) + (k ^ ((c & 15) << 2))] = W[idx];
    }
    __syncthreads();

    // ---- this wave's 16-row slab ----
    int slab = blockIdx.x * 8 + wv;       // 5000 slabs, 625 blocks exactly
    size_t row0 = (size_t)slab * 16;

    const float* aPtr = agg + (row0 + l16) * CIN + half * 2;
    int sw = l16 << 2;                    // per-lane swizzle term (bits >= 2)
    // half*2 only touches bit 1; k and sw only touch bits >= 2, so
    // (k + 2*half) ^ sw == (k ^ sw) + 2*half  -> fold half*2 into base.
    const float* wbase = &ldsWT[(l16 << 7) + half * 2];

    v8f acc[8];
    #pragma unroll
    for (int t = 0; t < 8; ++t) acc[t] = (v8f){};

    #pragma unroll 4
    for (int k = 0; k < CIN; k += 4) {
        // A fragment: one b64 per lane, reused by all 8 column tiles
        float2 av = *(const float2*)(aPtr + k);
        v2f a; a.x = av.x; a.y = av.y;

        // LOAD phase: 8 B fragments (4x ds_load_2addr) into distinct regs
        const float* wk = wbase + (k ^ sw);
        float2 frag[8];
        #pragma unroll
        for (int t = 0; t < 8; ++t)
            frag[t] = *(const float2*)(wk + t * 2048);   // ds b64, imm offsets

        // Hard phase split: nothing crosses, so all ds loads issue
        // before the first WMMA -> 4 live quads, staggered dscnt waits.
        __builtin_amdgcn_sched_barrier(0);

        // COMPUTE phase: 8 WMMAs, LDS latency hidden behind earlier ones
        #pragma unroll
        for (int t = 0; t < 8; ++t) {
            v2f bm; bm.x = frag[t].x; bm.y = frag[t].y;
            acc[t] = __builtin_amdgcn_wmma_f32_16x16x4_f32(
                         false, a, false, bm, (short)0, acc[t], false, false);
        }
    }

    // ---- bias + ReLU epilogue ----
    #pragma unroll
    for (int t = 0; t < 8; ++t) {
        float bv = bias[t * 16 + l16];
        #pragma unroll
        for (int m = 0; m < 8; ++m) {
            int r = m + half * 8;
            float val = acc[t][m] + bv;
            out[(row0 + r) * COUT + t * 16 + l16] = fmaxf(val, 0.f);
        }
    }
}

// ------------------------------------------------------------------
// Launch: zero agg -> SpMM atomics -> WMMA GEMM (all on `stream`).
// ------------------------------------------------------------------
extern "C" void kernel_launch(void* const* d_in, const int* in_sizes, int n_in,
                              void* d_out, int out_size, void* d_ws, size_t ws_size,
                              hipStream_t stream) {
    const float* x    = (const float*)d_in[0];   // [B,N,CIN] f32
    const int*   rows = (const int*)  d_in[1];   // [B,E] i32
    const int*   cols = (const int*)  d_in[2];   // [B,E] i32
    const float* vals = (const float*)d_in[3];   // [B,E] f32
    const float* W    = (const float*)d_in[4];   // [CIN,COUT] f32
    const float* bias = (const float*)d_in[5];   // [COUT] f32
    float*       out  = (float*)d_out;           // [B,N,COUT] f32
    float*       agg  = (float*)d_ws;            // B*N*CIN floats = 40.96 MB

    (void)in_sizes; (void)n_in; (void)out_size; (void)ws_size;

    // 1) zero accumulator
    int n4 = (BATCH * NNODE * CIN) / 4;
    gcn_zero_ws<<<(n4 + 255) / 256, 256, 0, stream>>>((float4*)agg, n4);

    // 2) SpMM: one wave per edge
    long spmmThreads = (long)BATCH * NEDGE * 32;
    int  spmmBlocks  = (int)((spmmThreads + 255) / 256);
    gcn_spmm_atomic<<<spmmBlocks, 256, 0, stream>>>(x, rows, cols, vals, agg);

    // 3) GEMM + bias + ReLU: 8 waves/block, one 16-row slab per wave,
    //    all 8 column tiles per wave. 5000 slabs -> exactly 625 blocks.
    gcn_gemm_wmma<<<625, 256, 0, stream>>>(agg, W, bias, out);
}